// Model_50354196578576
// MI455X (gfx1250) — compile-verified
//
#include <hip/hip_runtime.h>
#include <math.h>
#include <stdint.h>

// ---------------- problem constants ----------------
#define NN     102400          // nodes
#define GSZ    400             // nodes per block/group
#define NBLK   256             // NN / GSZ
#define DEG    16              // neighbors per node (plus 1 self loop)
#define ETOT   (NN * (DEG + 1))// 1,740,800 edges total (dst row offset)
#define FIN    128             // layer-0 input features
#define NH     4               // heads
#define DH     32              // per-head dim
#define KTOP   30              // sort-pool top-k
#define C1O    16
#define P1L    15              // K/2
#define C2O    32
#define C2L    11              // 15 - 5 + 1
#define OUTB   (C2O * C2L)     // 352 outputs per block

typedef float v2f __attribute__((ext_vector_type(2)));
typedef float v8f __attribute__((ext_vector_type(8)));
typedef unsigned int u32x4 __attribute__((ext_vector_type(4)));
typedef int i32x8 __attribute__((ext_vector_type(8)));
typedef int i32x4 __attribute__((ext_vector_type(4)));

// ---------------- LDS word offsets ----------------
//  bufA : 12800 | bufB : 12800 | hb : 12800 (reused as pooled/c1/p1)
//  Wlds : 4096  | s1 : 400 | s2 : 400 | dstl : 6400 ints | topk : 32 ints
#define OFF_A    0
#define OFF_B    12800
#define OFF_H    25600
#define OFF_W    38400
#define OFF_S1   42496
#define OFF_S2   42896
#define OFF_DST  43296
#define OFF_TOPK 49696
#define SMEM_WORDS 49728
#define SMEM_BYTES (SMEM_WORDS * 4)

// ---- Tensor Data Mover: 1-D DMA of `nelems` f32 global -> LDS (per-wave) ----
// D# per CDNA5 ISA: group0 = {count|ldsaddr|gaddr|type},
// group1 = {data_size=4B, tensor_dim0=nelems, tensor_dim1=1, tile=nelems x 1}.
__device__ __forceinline__ void tdm_load_1d(const float* gsrc,
                                            unsigned ldsByteOff,
                                            unsigned nelems) {
  const unsigned long long ga = (unsigned long long)(uintptr_t)gsrc;
  u32x4 g0;
  g0.x = 1u;                                               // count=1
  g0.y = ldsByteOff;                                       // lds_addr
  g0.z = (unsigned)ga;                                     // global_addr[31:0]
  g0.w = ((unsigned)(ga >> 32) & 0x01FFFFFFu) | 0x80000000u; // ga[56:32]|type=2
  i32x8 g1;
  g1[0] = 0x00020000;                                      // data_size = 4 bytes
  g1[1] = (int)((nelems & 0xFFFFu) << 16);                 // tensor_dim0 lo
  g1[2] = (int)((nelems >> 16) | (1u << 16));              // dim0 hi | dim1=1
  g1[3] = (int)((nelems & 0xFFFFu) << 16);                 // tile_dim0
  g1[4] = 1;                                               // tile_dim1 = 1
  g1[5] = (int)nelems;                                     // dim0_stride lo
  g1[6] = 0;
  g1[7] = 0;
  i32x4 z4 = {0, 0, 0, 0};
  i32x8 z8 = {0, 0, 0, 0, 0, 0, 0, 0};
#if __has_builtin(__builtin_amdgcn_tensor_load_to_lds)
  // 6-arg form (clang-23 / therock-10.0 headers)
  __builtin_amdgcn_tensor_load_to_lds(g0, g1, z4, z4, z8, 0);
#else
  asm volatile("tensor_load_to_lds %0, %1" ::"s"(g0), "s"(g1) : "memory");
#endif
  (void)z4; (void)z8;
}

__device__ __forceinline__ void wait_tensorcnt0() {
#if __has_builtin(__builtin_amdgcn_s_wait_tensorcnt)
  __builtin_amdgcn_s_wait_tensorcnt(0);
#else
  asm volatile("s_wait_tensorcnt 0x0" ::: "memory");
#endif
}

// 16x16 f32 WMMA tile GEMM: D[400x32] = A[400xFINT] * B[FINTx32].
// One wave per 16-row m-tile, two n-accumulators share each A fragment.
// B staged in LDS (Wlds) by the TDM.
template <bool FROM_GLOBAL, int FINT>
__device__ __forceinline__ void gemm_head(const float* __restrict__ Ag,
                                          const float* Al, const float* Wlds,
                                          float* hb, int wave, int lane) {
  const int lm   = lane & 15;
  const int half = lane >> 4;            // 0: K{0,1}, 1: K{2,3}
  for (int mt = wave; mt < 25; mt += 8) {
    const int m0  = mt << 4;
    const int row = m0 + lm;
    v8f acc0 = {0.f, 0.f, 0.f, 0.f, 0.f, 0.f, 0.f, 0.f};
    v8f acc1 = {0.f, 0.f, 0.f, 0.f, 0.f, 0.f, 0.f, 0.f};
#pragma unroll 8
    for (int k0 = 0; k0 < FINT; k0 += 4) {
      const int kk = k0 + (half << 1);
      v2f a, b0, b1;
      if (FROM_GLOBAL) {
        const float* ap = Ag + (size_t)row * FIN + kk;
        a.x = ap[0]; a.y = ap[1];
      } else {
        a.x = Al[row * DH + kk];
        a.y = Al[row * DH + kk + 1];
      }
      b0.x = Wlds[kk * DH + lm];
      b0.y = Wlds[(kk + 1) * DH + lm];
      b1.x = Wlds[kk * DH + 16 + lm];
      b1.y = Wlds[(kk + 1) * DH + 16 + lm];
      acc0 = __builtin_amdgcn_wmma_f32_16x16x4_f32(false, a, false, b0,
                                                   (short)0, acc0, false, false);
      acc1 = __builtin_amdgcn_wmma_f32_16x16x4_f32(false, a, false, b1,
                                                   (short)0, acc1, false, false);
    }
    const int rbase = m0 + (half ? 8 : 0);
#pragma unroll
    for (int r = 0; r < 8; ++r) {
      hb[(rbase + r) * DH + lm]      = acc0[r];
      hb[(rbase + r) * DH + 16 + lm] = acc1[r];
    }
  }
}

__global__ __launch_bounds__(256)
void gat_dgcnn_fused(const int* __restrict__ edge,      // [2, ETOT]
                     const float* __restrict__ xfeat,   // [NN, 128]
                     const float* __restrict__ W0,      // [4,128,32]
                     const float* __restrict__ a0,      // [4,64]
                     const float* __restrict__ Wl,      // [3,4,32,32]
                     const float* __restrict__ al,      // [3,4,64]
                     const float* __restrict__ w1,      // [16,1,128]
                     const float* __restrict__ b1,      // [16]
                     const float* __restrict__ w2,      // [32,16,5]
                     const float* __restrict__ b2,      // [32]
                     float* __restrict__ outp,          // [NBLK, 352]
                     float* __restrict__ cat01)         // ws: [NN, 64] (layers 0,1)
{
  extern __shared__ float sm[];
  float* bufA = sm + OFF_A;
  float* bufB = sm + OFF_B;
  float* hb   = sm + OFF_H;
  float* wlds = sm + OFF_W;
  float* s1   = sm + OFF_S1;
  float* s2   = sm + OFF_S2;
  int*   dstl = (int*)(sm + OFF_DST);
  int*   topk = (int*)(sm + OFF_TOPK);

  const int t    = threadIdx.x;
  const int wave = t >> 5;
  const int lane = t & 31;
  const int n0g  = blockIdx.x * GSZ;
  const int wq   = __builtin_amdgcn_readfirstlane(wave);  // wave-uniform SGPR

  // ---- warm L2 with this block's layer-0 rows (global_prefetch_b8) ----
  for (int p = t; p < GSZ; p += 256)
    __builtin_prefetch(xfeat + (size_t)(n0g + p) * FIN, 0, 1);

  // ---- preload block-local neighbor list (all edges stay in-block) ----
  const int* edst = edge + ETOT;                   // dst row
  for (int p = t; p < GSZ * DEG; p += 256)
    dstl[p] = edst[(size_t)n0g * DEG + p] - n0g;
  __syncthreads();

  // ---------------- 4 GAT layers ----------------
  for (int l = 0; l < 4; ++l) {
    float* xout = (l & 1) ? bufB : bufA;
    float* xin  = (l & 1) ? bufA : bufB;           // unused for l==0

    for (int p = t; p < GSZ * DH; p += 256) xout[p] = 0.f;
    __syncthreads();

    for (int h = 0; h < NH; ++h) {
      // ---- TDM: stage W[h] into LDS (each wave DMAs a 1/8 chunk) ----
      const float* Wg = (l == 0) ? (W0 + h * (FIN * DH))
                                 : (Wl + ((l - 1) * NH + h) * (DH * DH));
      const unsigned nel = (unsigned)((l == 0 ? FIN * DH : DH * DH) >> 3);
      tdm_load_1d(Wg + wq * nel, (unsigned)(OFF_W * 4) + wq * nel * 4u, nel);
      wait_tensorcnt0();
      __syncthreads();

      // ---- h = x @ W[h]  (WMMA f32 16x16x4) ----
      if (l == 0)
        gemm_head<true, FIN>(xfeat + (size_t)n0g * FIN, nullptr, wlds, hb,
                             wave, lane);
      else
        gemm_head<false, DH>(nullptr, xin, wlds, hb, wave, lane);
      __syncthreads();

      // ---- s1 = h . a[:32], s2 = h . a[32:] ----
      const float* av = (l == 0) ? (a0 + h * 2 * DH)
                                 : (al + ((l - 1) * NH + h) * 2 * DH);
      for (int i = t; i < GSZ; i += 256) {
        const float* hr = hb + i * DH;
        float d1 = 0.f, d2 = 0.f;
#pragma unroll
        for (int c = 0; c < DH; ++c) { d1 += hr[c] * av[c]; d2 += hr[c] * av[DH + c]; }
        s1[i] = d1; s2[i] = d2;
      }
      __syncthreads();

      // ---- edge-softmax aggregation: wave per node, lane per channel ----
      for (int i = wave; i < GSZ; i += 8) {
        const float s1i = s1[i];
        const int* dr = dstl + i * DEG;
        float rs = 0.f, acc = 0.f;
#pragma unroll 4
        for (int j = 0; j < DEG; ++j) {
          const int dl = dr[j];
          const float sl = s1i + s2[dl];
          const float lr = sl > 0.f ? sl : 0.2f * sl;
          const float e  = expf(-lr);
          rs  += e;
          acc += e * hb[dl * DH + lane];
        }
        { // self loop
          const float sl = s1i + s2[i];
          const float lr = sl > 0.f ? sl : 0.2f * sl;
          const float e  = expf(-lr);
          rs  += e;
          acc += e * hb[i * DH + lane];
        }
        float v = acc / rs;
        v = v > 0.f ? v : (expf(v) - 1.f);           // ELU
        xout[i * DH + lane] += 0.25f * v;            // mean over 4 heads
      }
      __syncthreads();
    }

    // layers 0,1 spill to ws (layers 2,3 stay live in LDS for the concat)
    if (l < 2) {
      for (int p = t; p < GSZ * DH; p += 256) {
        const int rrow = p >> 5, c = p & 31;
        cat01[(size_t)(n0g + rrow) * 64 + l * DH + c] = xout[p];
      }
      __threadfence();
      __syncthreads();
    }
  }
  // here: bufA = layer-2 out, bufB = layer-3 out; sort key = bufB[:,31]

  // ---------------- top-30 selection (wave 0, jax tie-break) ----------------
  if (t < 32) {
    float v[13];
#pragma unroll
    for (int q = 0; q < 13; ++q) {
      const int i = t + 32 * q;
      v[q] = (i < GSZ) ? bufB[i * DH + 31] : -INFINITY;
    }
    for (int it = 0; it < KTOP; ++it) {
      float bv = -INFINITY; int bi = 0x7fffffff;
#pragma unroll
      for (int q = 0; q < 13; ++q) {
        const int i = t + 32 * q;
        if (v[q] > bv || (v[q] == bv && i < bi)) { bv = v[q]; bi = i; }
      }
#pragma unroll
      for (int off = 16; off > 0; off >>= 1) {
        const float ov = __shfl_xor(bv, off, 32);
        const int   oi = __shfl_xor(bi, off, 32);
        if (ov > bv || (ov == bv && oi < bi)) { bv = ov; bi = oi; }
      }
      if (t == 0) topk[it] = bi;
      if ((bi & 31) == t) v[bi >> 5] = -INFINITY;   // mark selected
    }
  }
  __syncthreads();

  // ---------------- gather pooled [30 x 128] into LDS (reuse hb) ----------------
  float* pooled = hb;                 // 3840 floats
  float* c1b    = hb + KTOP * FIN;    // 480
  float* p1b    = c1b + C1O * KTOP;   // 240
  for (int p = t; p < KTOP * FIN; p += 256) {
    const int k = p >> 7, col = p & 127;
    const int node = topk[k];
    float val;
    if (col < 64)       val = cat01[(size_t)(n0g + node) * 64 + col];
    else if (col < 96)  val = bufA[node * DH + (col - 64)];
    else                val = bufB[node * DH + (col - 96)];
    pooled[p] = val;
  }
  __syncthreads();

  // ---------------- conv1 (1x128 pointwise) + relu ----------------
  for (int p = t; p < C1O * KTOP; p += 256) {
    const int o = p / KTOP, k = p % KTOP;
    float acc = b1[o];
#pragma unroll 8
    for (int d = 0; d < FIN; ++d) acc += pooled[k * FIN + d] * w1[o * FIN + d];
    c1b[o * KTOP + k] = acc > 0.f ? acc : 0.f;
  }
  __syncthreads();

  // ---------------- maxpool pairs over k ----------------
  for (int p = t; p < C1O * P1L; p += 256) {
    const int o = p / P1L, j = p % P1L;
    const float x0 = c1b[o * KTOP + 2 * j], x1 = c1b[o * KTOP + 2 * j + 1];
    p1b[o * P1L + j] = x0 > x1 ? x0 : x1;
  }
  __syncthreads();

  // ---------------- conv2 (16->32, width 5, valid) + relu ----------------
  for (int p = t; p < OUTB; p += 256) {
    const int oc = p / C2L, tt = p % C2L;
    float acc = b2[oc];
#pragma unroll
    for (int ic = 0; ic < C1O; ++ic) {
      const float* pr = p1b + ic * P1L + tt;
      const float* wr = w2 + (oc * C1O + ic) * 5;
#pragma unroll
      for (int r = 0; r < 5; ++r) acc += pr[r] * wr[r];
    }
    outp[(size_t)blockIdx.x * OUTB + p] = acc > 0.f ? acc : 0.f;
  }
}

extern "C" void kernel_launch(void* const* d_in, const int* in_sizes, int n_in,
                              void* d_out, int out_size, void* d_ws, size_t ws_size,
                              hipStream_t stream) {
  (void)in_sizes; (void)n_in; (void)out_size; (void)ws_size;
  const int*   edge  = (const int*)d_in[0];
  const float* xfeat = (const float*)d_in[1];
  const float* W0    = (const float*)d_in[2];
  const float* a0    = (const float*)d_in[3];
  const float* Wl    = (const float*)d_in[4];
  const float* al    = (const float*)d_in[5];
  const float* w1    = (const float*)d_in[6];
  const float* b1    = (const float*)d_in[7];
  const float* w2    = (const float*)d_in[8];
  const float* b2    = (const float*)d_in[9];
  float* outp  = (float*)d_out;
  float* cat01 = (float*)d_ws;      // NN * 64 floats = 26,214,400 B

  (void)hipFuncSetAttribute((const void*)gat_dgcnn_fused,
                            hipFuncAttributeMaxDynamicSharedMemorySize,
                            SMEM_BYTES);
  gat_dgcnn_fused<<<dim3(NBLK), dim3(256), SMEM_BYTES, stream>>>(
      edge, xfeat, W0, a0, Wl, al, w1, b1, w2, b2, outp, cat01);
}